// SpikeNeuralNetwork_79053168050675
// MI455X (gfx1250) — compile-verified
//
#include <hip/hip_runtime.h>

typedef float v2f __attribute__((ext_vector_type(2)));
typedef float v8f __attribute__((ext_vector_type(8)));
typedef float v3fa4 __attribute__((ext_vector_type(3), aligned(4)));
typedef unsigned int v2u __attribute__((ext_vector_type(2)));

#define T_STEPS 1000
#define BATCH   8192
#define IN_DIM  4
#define HID     10
#define OUT_DIM 3
#define BETA    0.95f
#define THR     1.0f

// v_permlane16_swap_b32: swaps lanes 16-31 of `a` with lanes 0-15 of `b`.
// Returns {new_a, new_b}:
//   new_a = { a[0:15],  b[0:15]  }   (b's low half moved into a's high half)
//   new_b = { a[16:31], b[16:31] }   (a's high half moved into b's low half)
struct SwapPair { float x, y; };
__device__ __forceinline__ SwapPair permswap(float a, float b) {
    v2u p = __builtin_amdgcn_permlane16_swap(__float_as_uint(a), __float_as_uint(b),
                                             false, false);
    SwapPair r; r.x = __uint_as_float(p.x); r.y = __uint_as_float(p.y);
    return r;
}

// One wave32 handles a tile of 16 batch elements for all 1000 timesteps.
// Layer1: cur1^T = W1(16x4) x x^T(4x16)  via V_WMMA_F32_16X16X4_F32
// Layer2: cur2^T = W2(16x10pad) x spk1^T(10x16) via 3 chained K=4 WMMAs.
__global__ __launch_bounds__(256) void snn_lif_wmma_kernel(
    const float* __restrict__ x,   // [T, B, IN]
    const float* __restrict__ W1,  // [HID, IN]
    const float* __restrict__ b1,  // [HID]
    const float* __restrict__ W2,  // [OUT, HID]
    const float* __restrict__ b2,  // [OUT]
    float* __restrict__ spk_out,   // [T, B, OUT]
    float* __restrict__ mem_out)   // [T, B, OUT]
{
    const int lane = threadIdx.x & 31;
    const int wave = threadIdx.x >> 5;
    const int tile = blockIdx.x * (blockDim.x >> 5) + wave;
    const int b0   = tile * 16;
    const bool lo  = (lane < 16);
    const int  bl  = lane & 15;          // matrix row (A/C) or batch column index
    const int  koff = lo ? 0 : 2;        // K split between lane halves

    // ---- loop-invariant matrix operands -------------------------------
    // A1 = W1, M=hid (padded to 16), K=4. Lane bl holds row hid=bl.
    v2f a1;
    a1.x = (bl < HID) ? W1[bl * IN_DIM + koff]     : 0.f;
    a1.y = (bl < HID) ? W1[bl * IN_DIM + koff + 1] : 0.f;

    // C1 init: b1 broadcast along batch columns (row = VGPR index (+8 hi half))
    v8f c1i;
#pragma unroll
    for (int v = 0; v < 8; ++v) {
        int h = v + (lo ? 0 : 8);
        c1i[v] = (h < HID) ? b1[h] : 0.f;
    }

    // A2 chunks = W2, M=out (padded to 16), K=4 per chunk (3 chunks cover K=10+pad).
    // Chunk 2 covers k=8..11; k=10,11 entries are ZERO, which also neutralizes the
    // junk rows left in the upper halves of the permlane16_swap outputs below.
    v2f a2[3];
#pragma unroll
    for (int c = 0; c < 3; ++c) {
        int k = 4 * c + koff;
        a2[c].x = (bl < OUT_DIM && k     < HID) ? W2[bl * HID + k]     : 0.f;
        a2[c].y = (bl < OUT_DIM && (k+1) < HID) ? W2[bl * HID + k + 1] : 0.f;
    }

    // C2 init: b2 broadcast along batch columns
    v8f c2i;
#pragma unroll
    for (int v = 0; v < 8; ++v) {
        int o = v + (lo ? 0 : 8);
        c2i[v] = (o < OUT_DIM) ? b2[o] : 0.f;
    }

    // ---- state --------------------------------------------------------
    float mem1[8];
#pragma unroll
    for (int v = 0; v < 8; ++v) mem1[v] = 0.f;
    float mem2[3] = {0.f, 0.f, 0.f};

    // x addressing: lane loads x[t][b0+bl][koff .. koff+1] (float2).
    // One tile+timestep = 256 contiguous bytes, fully coalesced per wave.
    const float* xbase = x + ((size_t)b0 + (size_t)bl) * IN_DIM + koff;
    const size_t xstride = (size_t)BATCH * IN_DIM;

    // 4-deep software pipeline on x loads to keep HBM requests in flight.
    v2f xq[4];
#pragma unroll
    for (int i = 0; i < 4; ++i)
        xq[i] = __builtin_nontemporal_load((const v2f*)(xbase + (size_t)i * xstride));

    for (int t = 0; t < T_STEPS; t += 4) {
#pragma unroll
        for (int j = 0; j < 4; ++j) {
            const int tc = t + j;
            v2f xb = xq[j];
            // prefetch t+j+4 (clamped; redundant tail loads are harmless)
            int tp = tc + 4; if (tp > T_STEPS - 1) tp = T_STEPS - 1;
            xq[j] = __builtin_nontemporal_load((const v2f*)(xbase + (size_t)tp * xstride));

            // ---- layer 1: cur1^T = A1 x xb + b1 -----------------------
            v8f d1 = __builtin_amdgcn_wmma_f32_16x16x4_f32(
                false, a1, false, xb, (short)0, c1i, false, false);

            // LIF layer 1 (reset uses mem_{t-1}; spike uses updated mem)
            float spk[8];
#pragma unroll
            for (int v = 0; v < 8; ++v) {
                float r = (mem1[v] > THR) ? THR : 0.f;
                mem1[v] = BETA * mem1[v] + (d1[v] - r);
                spk[v]  = (mem1[v] > THR) ? 1.f : 0.f;
            }

            // ---- build B chunks for layer 2 (spk1^T, K=4 slices) ------
            // spk[v]: row h = v (lanes 0-15) / v+8 (lanes 16-31), col = lane&15.
            // permlane16_swap pairs give both the k<8 chunks (.x) and the
            // k=8,9 chunk (.y) in one whole-wave VALU op each; junk rows
            // h=10,11 in .y's upper half are killed by zero A2 entries.
            SwapPair p02 = permswap(spk[0], spk[2]);
            SwapPair p13 = permswap(spk[1], spk[3]);
            SwapPair p46 = permswap(spk[4], spk[6]);
            SwapPair p57 = permswap(spk[5], spk[7]);
            v2f bb0, bb1, bb2;
            bb0.x = p02.x;  bb0.y = p13.x;   // k = 0,1 | 2,3
            bb1.x = p46.x;  bb1.y = p57.x;   // k = 4,5 | 6,7
            bb2.x = p02.y;  bb2.y = p13.y;   // k = 8,9 | (10,11 junk x 0)

            // ---- layer 2: three chained K=4 WMMAs ---------------------
            v8f d2 = __builtin_amdgcn_wmma_f32_16x16x4_f32(
                false, a2[0], false, bb0, (short)0, c2i, false, false);
            d2 = __builtin_amdgcn_wmma_f32_16x16x4_f32(
                false, a2[1], false, bb1, (short)0, d2, false, false);
            d2 = __builtin_amdgcn_wmma_f32_16x16x4_f32(
                false, a2[2], false, bb2, (short)0, d2, false, false);

            // LIF layer 2. d2[v], lanes 0-15: cur2 for out=v, batch=b0+lane.
#pragma unroll
            for (int o = 0; o < OUT_DIM; ++o) {
                float r = (mem2[o] > THR) ? THR : 0.f;
                mem2[o] = BETA * mem2[o] + (d2[o] - r);
            }

            // ---- stream results (lanes 0-15 own batch b0+lane) --------
            if (lo) {
                size_t idx = ((size_t)tc * BATCH + (size_t)(b0 + lane)) * OUT_DIM;
                v3fa4 sv, mv;
                sv.x = (mem2[0] > THR) ? 1.f : 0.f;
                sv.y = (mem2[1] > THR) ? 1.f : 0.f;
                sv.z = (mem2[2] > THR) ? 1.f : 0.f;
                mv.x = mem2[0]; mv.y = mem2[1]; mv.z = mem2[2];
                __builtin_nontemporal_store(sv, (v3fa4*)(spk_out + idx));
                __builtin_nontemporal_store(mv, (v3fa4*)(mem_out + idx));
            }
        }
    }
}

extern "C" void kernel_launch(void* const* d_in, const int* in_sizes, int n_in,
                              void* d_out, int out_size, void* d_ws, size_t ws_size,
                              hipStream_t stream) {
    (void)in_sizes; (void)n_in; (void)d_ws; (void)ws_size; (void)out_size;
    const float* x  = (const float*)d_in[0];
    const float* W1 = (const float*)d_in[1];
    const float* b1 = (const float*)d_in[2];
    const float* W2 = (const float*)d_in[3];
    const float* b2 = (const float*)d_in[4];
    float* spk = (float*)d_out;                                   // [T,B,OUT]
    float* mem = spk + (size_t)T_STEPS * BATCH * OUT_DIM;         // [T,B,OUT]

    // 8192 / 16 = 512 tiles -> 512 waves -> 64 blocks x 256 threads (8 waves)
    dim3 grid(BATCH / 16 / 8), block(256);
    snn_lif_wmma_kernel<<<grid, block, 0, stream>>>(x, W1, b1, W2, b2, spk, mem);
}